// SATCausalAttention_61735859912743
// MI455X (gfx1250) — compile-verified
//
#include <hip/hip_runtime.h>
#include <math.h>

// ---------------------------------------------------------------------------
// SAT causal attention + GNN structural bias, CDNA5 (gfx1250, wave32).
// All GEMMs use V_WMMA_F32_16X16X4_F32 (full fp32 precision, matrix pipe).
// Wave = 32x32 macro-tile (2x2 WMMA tiles): 4 WMMAs per 4 LDS pair-loads.
// K-chunks are double-buffered in LDS; next chunk's global loads are issued
// before the current 32-WMMA burst so HBM latency overlaps the matrix pipe.
// B tile is stored K-contiguous (Bs[n][k]) so every fragment is a single
// ds_load_2addr of consecutive dwords (no register re-pairing movs).
// ---------------------------------------------------------------------------

typedef __attribute__((ext_vector_type(2))) float v2f;
typedef __attribute__((ext_vector_type(8))) float v8f;

#define BATCH 4
#define SEQ   1024
#define DMODEL 512
#define NH    8
#define HD    64
#define EDG   65536
#define NTOK  (BATCH*SEQ)     // 4096
#define FBN   (2*DMODEL)      // 1024 batchnorm features
#define BN_EPS 1e-5f

// ---------------------------------------------------------------------------
// Generic fp32 WMMA GEMM: C[z] = alpha * A[z] @ B[z] (+bias) (+residual A row)
// Block: 256 threads = 8 waves (4 M x 2 N), computes a 128(M) x 64(N) tile.
// Per-z base: ptr + (z/zdiv)*s1 + (z%zdiv)*s2  (covers [B,H,...] layouts).
// Flags: GATHER_A: A row m comes from A[gidx[m]]
//        TRANS_B : B operand stored [N,K] row-major (computes A @ B^T)
//        SCATTER_C: atomicAdd C rows into row sidx[m] (segment-sum fusion)
//        RESIDUAL: add A[row][n] to the output (requires N==K semantics)
// M % 128 == 0, N % 64 == 0, K % 32 == 0 for every call in this problem.
// ---------------------------------------------------------------------------
template<bool GATHER_A, bool TRANS_B, bool SCATTER_C, bool RESIDUAL>
__global__ __launch_bounds__(256)
void gemm_wmma_f32(const float* __restrict__ A, long long sA1, long long sA2, int lda,
                   const float* __restrict__ Bm, long long sB1, long long sB2, int ldb,
                   float* __restrict__ C, long long sC1, long long sC2, int ldc,
                   const float* __restrict__ bias, float alpha,
                   int M, int N, int K, int zdiv,
                   const int* __restrict__ gidx, const int* __restrict__ sidx)
{
    __shared__ float As[2][128][33];   // 33.8 KB  (A: [m][k])
    __shared__ float Bs[2][64][33];    // 16.9 KB  (B: [n][k], K-contiguous)

    const int z  = blockIdx.z;
    const long long zq = z / zdiv, zr = z % zdiv;
    const float* Ab = A  + zq * sA1 + zr * sA2;
    const float* Bb = Bm + zq * sB1 + zr * sB2;
    float*       Cb = C  + zq * sC1 + zr * sC2;

    const int n0   = blockIdx.x * 64;
    const int m0   = blockIdx.y * 128;
    const int tid  = threadIdx.x;
    const int lane = tid & 31;
    const int wave = tid >> 5;        // 0..7
    const int wm   = wave >> 1;       // 0..3  (M direction, 32 rows each)
    const int wn   = wave & 1;        // 0..1  (N direction, 32 cols each)
    const int hl   = lane >> 4;       // lane half (K-pair / M-half select)
    const int l15  = lane & 15;

    v8f acc[2][2] = {};

    // ---- staging coordinates (fixed across K chunks) -----------------------
    const int ar  = tid >> 1;                 // A row within tile (128 rows)
    const int ac0 = (tid & 1) * 16;           // 16 floats per thread
    long long arowg = m0 + ar;
    if (GATHER_A) arowg = gidx[arowg];
    const float* asrc0 = Ab + arowg * (long long)lda + ac0;

    int btn, btk;                             // B staging coords
    const float* bsrc0;
    if (TRANS_B) {                            // B[N,K]: row copy into Bs[n][k]
        btn = tid >> 2;                       // 0..63  (N)
        btk = (tid & 3) * 8;                  // 0..24  (K)
        bsrc0 = Bb + (long long)(n0 + btn) * ldb + btk;
    } else {                                  // B[K,N]: transpose into Bs[n][k]
        btk = tid >> 3;                       // 0..31  (K)
        btn = (tid & 7) * 8;                  // 0..56  (N)
        bsrc0 = Bb + (long long)btk * ldb + n0 + btn;
    }

    float4 ra0, ra1, ra2, ra3, rb0, rb1;

    // global -> registers for chunk starting at k0
    auto g2r = [&](int k0) {
        const float* s = asrc0 + k0;
        ra0 = *(const float4*)(s);
        ra1 = *(const float4*)(s + 4);
        ra2 = *(const float4*)(s + 8);
        ra3 = *(const float4*)(s + 12);
        const float* t = TRANS_B ? (bsrc0 + k0) : (bsrc0 + (long long)k0 * ldb);
        rb0 = *(const float4*)(t);
        rb1 = *(const float4*)(t + 4);
    };
    // registers -> LDS buffer
    auto r2s = [&](int buf) {
        float* a = &As[buf][ar][ac0];
        a[ 0]=ra0.x; a[ 1]=ra0.y; a[ 2]=ra0.z; a[ 3]=ra0.w;
        a[ 4]=ra1.x; a[ 5]=ra1.y; a[ 6]=ra1.z; a[ 7]=ra1.w;
        a[ 8]=ra2.x; a[ 9]=ra2.y; a[10]=ra2.z; a[11]=ra2.w;
        a[12]=ra3.x; a[13]=ra3.y; a[14]=ra3.z; a[15]=ra3.w;
        if (TRANS_B) {                         // contiguous along K
            float* b = &Bs[buf][btn][btk];
            b[0]=rb0.x; b[1]=rb0.y; b[2]=rb0.z; b[3]=rb0.w;
            b[4]=rb1.x; b[5]=rb1.y; b[6]=rb1.z; b[7]=rb1.w;
        } else {                               // scatter across N rows
            Bs[buf][btn+0][btk]=rb0.x; Bs[buf][btn+1][btk]=rb0.y;
            Bs[buf][btn+2][btk]=rb0.z; Bs[buf][btn+3][btk]=rb0.w;
            Bs[buf][btn+4][btk]=rb1.x; Bs[buf][btn+5][btk]=rb1.y;
            Bs[buf][btn+6][btk]=rb1.z; Bs[buf][btn+7][btk]=rb1.w;
        }
    };

    const int am0 = wm * 32 + l15, am1 = am0 + 16;
    const int bn0 = wn * 32 + l15, bn1 = bn0 + 16;
    const int nch = K >> 5;

    g2r(0);
    r2s(0);
    __syncthreads();

    for (int c = 0; c < nch; ++c) {
        const int cur = c & 1;
        if (c + 1 < nch) g2r((c + 1) << 5);    // overlap HBM with WMMA burst

        // ---- 8 K-steps x 4 WMMA (2x2 macro-tile) --------------------------
        // A frag (16x4): lanes 0-15 hold {K,K+1} of row M=lane,
        //                lanes 16-31 hold {K+2,K+3} (ISA 7.12.2).
        #pragma unroll
        for (int kk = 0; kk < 32; kk += 4) {
            const int k2 = kk + 2 * hl;
            v2f af0, af1, bf0, bf1;
            af0.x = As[cur][am0][k2];  af0.y = As[cur][am0][k2+1];
            af1.x = As[cur][am1][k2];  af1.y = As[cur][am1][k2+1];
            bf0.x = Bs[cur][bn0][k2];  bf0.y = Bs[cur][bn0][k2+1];
            bf1.x = Bs[cur][bn1][k2];  bf1.y = Bs[cur][bn1][k2+1];
            acc[0][0] = __builtin_amdgcn_wmma_f32_16x16x4_f32(
                          false, af0, false, bf0, (short)0, acc[0][0], false, false);
            acc[0][1] = __builtin_amdgcn_wmma_f32_16x16x4_f32(
                          false, af0, false, bf1, (short)0, acc[0][1], false, false);
            acc[1][0] = __builtin_amdgcn_wmma_f32_16x16x4_f32(
                          false, af1, false, bf0, (short)0, acc[1][0], false, false);
            acc[1][1] = __builtin_amdgcn_wmma_f32_16x16x4_f32(
                          false, af1, false, bf1, (short)0, acc[1][1], false, false);
        }

        if (c + 1 < nch) {
            r2s((c + 1) & 1);
            __syncthreads();
        }
    }

    // ---- epilogue: C layout VGPR r -> M = r + 8*hl, N = lane&15 ------------
    #pragma unroll
    for (int tm = 0; tm < 2; ++tm) {
        #pragma unroll
        for (int tn = 0; tn < 2; ++tn) {
            const int gcol = n0 + wn * 32 + tn * 16 + l15;
            #pragma unroll
            for (int r = 0; r < 8; ++r) {
                const int grow = m0 + wm * 32 + tm * 16 + r + 8 * hl;
                float val = alpha * acc[tm][tn][r];
                if (bias) val += bias[gcol];
                if (RESIDUAL) {
                    long long arow = GATHER_A ? (long long)gidx[grow] : (long long)grow;
                    val += Ab[arow * (long long)lda + gcol];
                }
                if (SCATTER_C) {
                    atomicAdd(Cb + (long long)sidx[grow] * ldc + gcol, val);
                } else {
                    Cb[(long long)grow * ldc + gcol] = val;
                }
            }
        }
    }
}

// ---------------------------------------------------------------------------
// Fused bias + causal mask + softmax (in place over scores in d_out).
// scores[bh,q,k] += ss*Kst[b,q,k] + Bg[b,q,k]; masked (k>q) -> exactly -10000.
// ---------------------------------------------------------------------------
__global__ __launch_bounds__(256)
void softmax_bias_kernel(float* __restrict__ attn,
                         const float* __restrict__ Kst,
                         const float* __restrict__ Bg,
                         const float* __restrict__ ss_ptr)
{
    const int row = blockIdx.x;            // 0 .. B*H*L-1
    const int q   = row % SEQ;
    const int bh  = row / SEQ;
    const int b   = bh / NH;
    const float ss = ss_ptr[0];

    float*       srow = attn + (size_t)bh * SEQ * SEQ + (size_t)q * SEQ;
    const float* krow = Kst  + ((size_t)b * SEQ + q) * SEQ;
    const float* grow = Bg   + ((size_t)b * SEQ + q) * SEQ;

    const int tid = threadIdx.x;
    float vals[4];
    float mx = -3.0e38f;
    #pragma unroll
    for (int i = 0; i < 4; ++i) {
        const int kc = tid + i * 256;
        float v = srow[kc] + ss * krow[kc] + grow[kc];
        if (kc > q) v = -10000.0f;          // reference overwrites, not -inf
        vals[i] = v;
        mx = fmaxf(mx, v);
    }
    __shared__ float redm[8], reds[8];
    for (int o = 16; o > 0; o >>= 1) mx = fmaxf(mx, __shfl_xor(mx, o, 32));
    if ((tid & 31) == 0) redm[tid >> 5] = mx;
    __syncthreads();
    mx = redm[0];
    #pragma unroll
    for (int w = 1; w < 8; ++w) mx = fmaxf(mx, redm[w]);

    float s = 0.f;
    #pragma unroll
    for (int i = 0; i < 4; ++i) { vals[i] = expf(vals[i] - mx); s += vals[i]; }
    for (int o = 16; o > 0; o >>= 1) s += __shfl_xor(s, o, 32);
    if ((tid & 31) == 0) reds[tid >> 5] = s;
    __syncthreads();
    s = 0.f;
    #pragma unroll
    for (int w = 0; w < 8; ++w) s += reds[w];
    const float inv = 1.0f / s;
    #pragma unroll
    for (int i = 0; i < 4; ++i) srow[tid + i * 256] = vals[i] * inv;
}

// ---------------------------------------------------------------------------
// Small elementwise / reduction helpers
// ---------------------------------------------------------------------------
__global__ void count_kernel(const int* __restrict__ sind, float* __restrict__ cnt)
{
    const int e = blockIdx.x * blockDim.x + threadIdx.x;
    if (e < EDG) atomicAdd(&cnt[sind[e]], 1.0f);
}

__global__ void divide_kernel(float* __restrict__ pool, const float* __restrict__ cnt)
{
    const int i = blockIdx.x * blockDim.x + threadIdx.x;   // N*D elements
    const int r = i >> 9;                                   // / 512
    pool[i] = pool[i] / fmaxf(cnt[r], 1.0f);
}

__global__ void bn_partial(const float* __restrict__ x, const float* __restrict__ pool,
                           float* __restrict__ csum, float* __restrict__ csq)
{
    const int col = blockIdx.x * 256 + threadIdx.x;   // 0..FBN-1
    const int r0  = blockIdx.y * 256;
    float s = 0.f, s2 = 0.f;
    for (int r = r0; r < r0 + 256; ++r) {
        const float v = (col < DMODEL) ? x[(size_t)r * DMODEL + col]
                                       : pool[(size_t)r * DMODEL + (col - DMODEL)];
        s += v; s2 += v * v;
    }
    atomicAdd(&csum[col], s);
    atomicAdd(&csq[col], s2);
}

__global__ void bn_finalize(const float* __restrict__ csum, const float* __restrict__ csq,
                            float* __restrict__ mu, float* __restrict__ rs)
{
    const int c = blockIdx.x * 256 + threadIdx.x;
    if (c < FBN) {
        const float m = csum[c] * (1.0f / NTOK);
        const float v = csq[c] * (1.0f / NTOK) - m * m;     // biased var (jnp.var)
        mu[c] = m;
        rs[c] = rsqrtf(v + BN_EPS);
    }
}

__global__ void normalize_kernel(const float* __restrict__ x, const float* __restrict__ pool,
                                 const float* __restrict__ mu, const float* __restrict__ rs,
                                 const float* __restrict__ gamma, const float* __restrict__ beta,
                                 float* __restrict__ nrm)
{
    const int i = blockIdx.x * blockDim.x + threadIdx.x;   // N*FBN elements
    const int r = i >> 10;
    const int c = i & (FBN - 1);
    const float v = (c < DMODEL) ? x[(size_t)r * DMODEL + c]
                                 : pool[(size_t)r * DMODEL + (c - DMODEL)];
    nrm[i] = (v - mu[c]) * rs[c] * gamma[c] + beta[c];
}

// ---------------------------------------------------------------------------
// Host-side pipeline
// ---------------------------------------------------------------------------
extern "C" void kernel_launch(void* const* d_in, const int* in_sizes, int n_in,
                              void* d_out, int out_size, void* d_ws, size_t ws_size,
                              hipStream_t stream)
{
    (void)in_sizes; (void)n_in; (void)out_size; (void)ws_size;

    const float* x    = (const float*)d_in[0];
    // d_in[1] lattice: unused (cond_dim = 0)
    const float* Bg   = (const float*)d_in[2];
    const int*   sidx = (const int*)d_in[3];
    const int*   sind = (const int*)d_in[4];
    // d_in[5] attn_mask: deterministic causal triu(k=1), computed analytically
    const float* Wq = (const float*)d_in[6];   const float* bq = (const float*)d_in[7];
    const float* Wk = (const float*)d_in[8];   const float* bk = (const float*)d_in[9];
    const float* Wv = (const float*)d_in[10];  const float* bv = (const float*)d_in[11];
    const float* Wo = (const float*)d_in[12];  const float* bo = (const float*)d_in[13];
    const float* Wmp = (const float*)d_in[14]; const float* bmp = (const float*)d_in[15];
    const float* gamma = (const float*)d_in[16];
    const float* beta  = (const float*)d_in[17];
    const float* Wex = (const float*)d_in[18]; const float* bex = (const float*)d_in[19];
    const float* ss  = (const float*)d_in[20];

    float* out  = (float*)d_out;                       // [N, D]
    float* attn = out + (size_t)NTOK * DMODEL;         // [B*H, L, L] scores->attn

    // workspace layout (floats); total ~20.1M floats ~= 80.5 MB
    float* ws   = (float*)d_ws;
    const size_t M1 = 1u << 20;
    float* q    = ws;             // 2M  [N, H*HD]
    float* k    = ws + 2*M1;      // 2M
    float* v    = ws + 4*M1;      // 2M
    float* pool = ws + 6*M1;      // 2M  segment sums -> h_pool (in place)
    float* ex   = ws + 8*M1;      // 2M  x_struct
    float* head = ws + 10*M1;     // 2M  attn@v result [N, H*HD]
    float* Kst  = ws + 12*M1;     // 4M  [B, L, L]
    float* nrm  = ws + 16*M1;     // 4M  [N, 2D] batchnorm output
    float* cnt  = ws + 20*M1;     // 4096
    float* csum = cnt + 4096;     // 1024
    float* csq  = csum + 1024;    // 1024
    float* mu   = csq + 1024;     // 1024
    float* rs   = mu + 1024;      // 1024

    const dim3 blk(256);
    const long long LD = (long long)SEQ * DMODEL;      // per-batch token stride
    const long long LL = (long long)SEQ * SEQ;

    // zero accumulation buffers (graph-capturable memset nodes)
    hipMemsetAsync(pool, 0, (size_t)NTOK * DMODEL * sizeof(float), stream);
    hipMemsetAsync(cnt,  0, (4096 + 1024 + 1024) * sizeof(float), stream);

    // --- QKV projections: [4096,512] @ [512,512] + b ------------------------
    gemm_wmma_f32<false,false,false,false><<<dim3(8,32,1), blk, 0, stream>>>(
        x,0,0,DMODEL, Wq,0,0,DMODEL, q,0,0,DMODEL, bq,1.0f,
        NTOK,DMODEL,DMODEL, 1, nullptr,nullptr);
    gemm_wmma_f32<false,false,false,false><<<dim3(8,32,1), blk, 0, stream>>>(
        x,0,0,DMODEL, Wk,0,0,DMODEL, k,0,0,DMODEL, bk,1.0f,
        NTOK,DMODEL,DMODEL, 1, nullptr,nullptr);
    gemm_wmma_f32<false,false,false,false><<<dim3(8,32,1), blk, 0, stream>>>(
        x,0,0,DMODEL, Wv,0,0,DMODEL, v,0,0,DMODEL, bv,1.0f,
        NTOK,DMODEL,DMODEL, 1, nullptr,nullptr);

    // --- GNN extractor ------------------------------------------------------
    // h = x[sub_idx] + x[sub_idx]@Wmp + bmp, scatter-summed into pool[sub_ind]
    gemm_wmma_f32<true,false,true,true><<<dim3(8, EDG/128, 1), blk, 0, stream>>>(
        x,0,0,DMODEL, Wmp,0,0,DMODEL, pool,0,0,DMODEL, bmp,1.0f,
        EDG,DMODEL,DMODEL, 1, sidx, sind);
    count_kernel<<<EDG/256, blk, 0, stream>>>(sind, cnt);
    divide_kernel<<<(NTOK*DMODEL)/256, blk, 0, stream>>>(pool, cnt);

    // BatchNorm over concat([x, h_pool]) [4096, 1024]
    bn_partial<<<dim3(FBN/256, 16), blk, 0, stream>>>(x, pool, csum, csq);
    bn_finalize<<<FBN/256, blk, 0, stream>>>(csum, csq, mu, rs);
    normalize_kernel<<<(NTOK*FBN)/256, blk, 0, stream>>>(x, pool, mu, rs, gamma, beta, nrm);

    // ex = nrm @ Wex + bex  : [4096,1024] @ [1024,512]
    gemm_wmma_f32<false,false,false,false><<<dim3(8,32,1), blk, 0, stream>>>(
        nrm,0,0,FBN, Wex,0,0,DMODEL, ex,0,0,DMODEL, bex,1.0f,
        NTOK,DMODEL,FBN, 1, nullptr,nullptr);

    // --- structural Gram: Kst[b] = ex_b @ ex_b^T / sqrt(D) ------------------
    gemm_wmma_f32<false,true,false,false><<<dim3(16,8,BATCH), blk, 0, stream>>>(
        ex,LD,0,DMODEL, ex,LD,0,DMODEL, Kst,LL,0,SEQ, nullptr,
        1.0f/sqrtf((float)DMODEL), SEQ,SEQ,DMODEL, 1, nullptr,nullptr);

    // --- attention scores: q @ k^T / sqrt(DH), per (b,h) --------------------
    gemm_wmma_f32<false,true,false,false><<<dim3(16,8,BATCH*NH), blk, 0, stream>>>(
        q,LD,HD,DMODEL, k,LD,HD,DMODEL, attn,(long long)NH*LL,LL,SEQ, nullptr,
        0.125f, SEQ,SEQ,HD, NH, nullptr,nullptr);

    // --- fused struct bias + B_graph + causal mask + softmax (in place) -----
    softmax_bias_kernel<<<BATCH*NH*SEQ, blk, 0, stream>>>(attn, Kst, Bg, ss);

    // --- out heads: attn @ v ------------------------------------------------
    gemm_wmma_f32<false,false,false,false><<<dim3(1,8,BATCH*NH), blk, 0, stream>>>(
        attn,(long long)NH*LL,LL,SEQ, v,LD,HD,DMODEL, head,LD,HD,DMODEL, nullptr,
        1.0f, SEQ,HD,SEQ, NH, nullptr,nullptr);

    // --- output projection: head @ Wo + bo ---------------------------------
    gemm_wmma_f32<false,false,false,false><<<dim3(8,32,1), blk, 0, stream>>>(
        head,0,0,DMODEL, Wo,0,0,DMODEL, out,0,0,DMODEL, bo,1.0f,
        NTOK,DMODEL,DMODEL, 1, nullptr,nullptr);
}